// LinearCategoricalEncoding_80324478370074
// MI455X (gfx1250) — compile-verified
//
#include <hip/hip_runtime.h>
#include <math.h>

typedef __attribute__((ext_vector_type(2))) float v2f;
typedef __attribute__((ext_vector_type(8))) float v8f;

#define B_   64
#define S_   512
#define K_   64
#define D_   16
#define N_   (B_ * S_)
#define LOG2PI_F 1.8378770664093453f
#define XPAD 36            // padded row stride (floats) for the 16x32 X tile -> conflict-free b64 reads
#define PTS_PER_BLK 128    // 8 waves x one 16-point tile each
#define NBLK (N_ / PTS_PER_BLK)   // 256 blocks -> fills the WGP array

__global__ __launch_bounds__(256) void lce_kernel(
    const int*   __restrict__ z,      // [N] categories
    const float* __restrict__ zn,     // [N,16] z_noise
    const float* __restrict__ tg,     // [K,16] t
    const float* __restrict__ lsg,    // [K,16] log_s
    float* __restrict__ z_out,        // [N,16]
    float* __restrict__ ldj_part)     // [NBLK] partial row sums
{
    __shared__ float t_lds[K_ * D_];      // t[k][d]
    __shared__ float texp[K_ * D_];       // exp(log_s[k][d])
    __shared__ float Wt[32 * K_];         // Wt[j][k]: j<16 -> e^{-2s}, j>=16 -> -2 t e^{-s}
    __shared__ float Ek[K_];              // -0.5*sum t^2 - 8 ln(2pi) - sum s + cat_prior
    __shared__ float ssum[K_];            // sum_d log_s[k][d]
    __shared__ float Xs[8 * 16 * XPAD];   // per-wave X staging tile (16 pts x 32 cols)
    __shared__ float meta_lp[8 * 16];     // per-wave per-point log_point
    __shared__ int   meta_c[8 * 16];      // per-wave per-point category
    __shared__ float sh_ldj;

    const int   tid       = threadIdx.x;
    const float cat_prior = -logf((float)K_);

    // ---- build per-category tables -------------------------------------
    for (int idx = tid; idx < K_ * D_; idx += 256) {
        float sv = lsg[idx];
        float tv = tg[idx];
        int   k  = idx >> 4;          // category
        int   d  = idx & 15;
        t_lds[idx] = tv;
        texp[idx]  = __expf(sv);
        float ens  = __expf(-sv);
        Wt[d * K_ + k]        = ens * ens;        // e^{-2s}
        Wt[(16 + d) * K_ + k] = -2.0f * tv * ens; // -2 t e^{-s}
    }
    if (tid < K_) {
        float c2 = 0.0f, ss = 0.0f;
        for (int d = 0; d < D_; ++d) {
            float tv = tg[tid * D_ + d];
            float sv = lsg[tid * D_ + d];
            c2 += tv * tv;
            ss += sv;
        }
        ssum[tid] = ss;
        Ek[tid]   = -0.5f * c2 - 8.0f * LOG2PI_F - ss + cat_prior;
    }
    if (tid == 0) sh_ldj = 0.0f;
    __syncthreads();

    const int wave   = tid >> 5;
    const int lane   = tid & 31;
    const int laneLo = lane & 15;
    const int hi     = lane >> 4;     // 0: lanes 0-15, 1: lanes 16-31

    // ---- preload B operands (block-invariant) into registers ------------
    // B tile [4 x 16]: vgpr0 = row (0|2), vgpr1 = row (1|3); col = lane%16
    v2f breg[4][8];
    #pragma unroll
    for (int ct = 0; ct < 4; ++ct) {
        #pragma unroll
        for (int kk = 0; kk < 8; ++kk) {
            int row = 4 * kk + 2 * hi;
            int cat = ct * 16 + laneLo;
            breg[ct][kk].x = Wt[row * K_ + cat];
            breg[ct][kk].y = Wt[(row + 1) * K_ + cat];
        }
    }

    float  wave_ldj = 0.0f;
    float* xw       = &Xs[wave * 16 * XPAD];
    const int tbase = blockIdx.x * PTS_PER_BLK + wave * 16;  // this wave's 16 points

    // ---- staging: 2 lanes per point, 8 d's each -------------------------
    {
        const int ptl  = lane >> 1;
        const int half = lane & 1;
        const int p    = tbase + ptl;
        const int c    = z[p];
        const float* xr = zn + (size_t)p * D_ + half * 8;

        float zs[8];
        float sx2 = 0.0f;
        #pragma unroll
        for (int dd = 0; dd < 8; ++dd) {
            float x = xr[dd];
            sx2 += x * x;
            int d = half * 8 + dd;
            zs[dd] = (x + t_lds[c * D_ + d]) * texp[c * D_ + d];
        }
        sx2 += __shfl_xor(sx2, 1, 32);   // pair sum -> full sum_d x^2

        float* zo = z_out + (size_t)p * D_ + half * 8;
        #pragma unroll
        for (int dd = 0; dd < 8; ++dd) {
            int d = half * 8 + dd;
            xw[ptl * XPAD + d]      = zs[dd] * zs[dd];  // quadratic part
            xw[ptl * XPAD + 16 + d] = zs[dd];           // linear part
            zo[dd] = zs[dd];                            // z_enc output
        }
        if (half == 0) {
            meta_c[wave * 16 + ptl]  = c;
            meta_lp[wave * 16 + ptl] = -0.5f * sx2 - 8.0f * LOG2PI_F - ssum[c] + cat_prior;
        }
    }
    // DS ops from this wave are in-order; wait for our own LDS writes.
    asm volatile("s_wait_dscnt 0" ::: "memory");

    // ---- A operands: lane<16 cols {4kk,4kk+1}, lane>=16 cols {+2,+3} ----
    v2f areg[8];
    #pragma unroll
    for (int kk = 0; kk < 8; ++kk) {
        const float* ap = &xw[laneLo * XPAD + 4 * kk + 2 * hi];
        areg[kk].x = ap[0];
        areg[kk].y = ap[1];
    }

    // ---- fp32 WMMA: P[16pts x 64cats] = X(16x32) * Wt(32x64) ------------
    v8f acc[4];
    #pragma unroll
    for (int ct = 0; ct < 4; ++ct) {
        acc[ct] = (v8f){0.f, 0.f, 0.f, 0.f, 0.f, 0.f, 0.f, 0.f};
        #pragma unroll
        for (int kk = 0; kk < 8; ++kk) {
            acc[ct] = __builtin_amdgcn_wmma_f32_16x16x4_f32(
                false, areg[kk], false, breg[ct][kk],
                (short)0, acc[ct], false, false);
        }
    }

    // ---- epilogue: denom, true-cat swap, logsumexp over 64 cats ---------
    #pragma unroll
    for (int r = 0; r < 8; ++r) {
        const int   m  = r + hi * 8;                // point index in tile
        const int   cm = meta_c[wave * 16 + m];
        const float lp = meta_lp[wave * 16 + m];

        float dv[4];
        #pragma unroll
        for (int ct = 0; ct < 4; ++ct) {
            float v = -0.5f * acc[ct][r] + Ek[ct * 16 + laneLo];
            if (cm == ct * 16 + laneLo) v = lp;     // exact point prob swap
            dv[ct] = v;
        }
        float mx = fmaxf(fmaxf(dv[0], dv[1]), fmaxf(dv[2], dv[3]));
        #pragma unroll
        for (int off = 8; off >= 1; off >>= 1)
            mx = fmaxf(mx, __shfl_xor(mx, off, 16));
        float sm = 0.0f;
        #pragma unroll
        for (int ct = 0; ct < 4; ++ct) sm += __expf(dv[ct] - mx);
        #pragma unroll
        for (int off = 8; off >= 1; off >>= 1)
            sm += __shfl_xor(sm, off, 16);
        const float log_den = mx + __logf(sm);
        if (laneLo == 0) wave_ldj += (cat_prior - log_den);  // ldj_loc
    }

    // ---- block partial-sum of ldj (LDS float atomics, order not fixed; no global atomics) ----
    wave_ldj += __shfl_xor(wave_ldj, 16, 32);   // combine lanes 0 and 16
    if (lane == 0) atomicAdd(&sh_ldj, wave_ldj);
    __syncthreads();
    if (tid == 0) ldj_part[blockIdx.x] = sh_ldj;
}

// Deterministic second stage: 4 partials (one per quarter-row) -> ldj[b]
__global__ __launch_bounds__(64) void ldj_reduce_kernel(
    const float* __restrict__ ldj_part, float* __restrict__ ldj)
{
    const int b = threadIdx.x;
    if (b < B_) {
        ldj[b] = ((ldj_part[4 * b + 0] + ldj_part[4 * b + 1]) +
                  (ldj_part[4 * b + 2] + ldj_part[4 * b + 3]));
    }
}

extern "C" void kernel_launch(void* const* d_in, const int* in_sizes, int n_in,
                              void* d_out, int out_size, void* d_ws, size_t ws_size,
                              hipStream_t stream) {
    (void)in_sizes; (void)n_in; (void)out_size; (void)ws_size;
    const int*   z  = (const int*)d_in[0];
    const float* zn = (const float*)d_in[1];
    const float* tg = (const float*)d_in[2];
    const float* ls = (const float*)d_in[3];
    float* z_out    = (float*)d_out;
    float* ldj      = z_out + (size_t)N_ * D_;
    float* ldj_part = (float*)d_ws;             // NBLK floats of scratch

    hipLaunchKernelGGL(lce_kernel, dim3(NBLK), dim3(256), 0, stream,
                       z, zn, tg, ls, z_out, ldj_part);
    hipLaunchKernelGGL(ldj_reduce_kernel, dim3(1), dim3(64), 0, stream,
                       ldj_part, ldj);
}